// GINConv_69123203662130
// MI455X (gfx1250) — compile-verified
//
#include <hip/hip_runtime.h>
#include <hip/hip_bf16.h>

typedef float v2f __attribute__((ext_vector_type(2)));
typedef float v8f __attribute__((ext_vector_type(8)));

#define D 128
#define WAVES_PER_BLOCK 8

// ---------------------------------------------------------------------------
// Kernel 1: h0 = X   (eps = 0  =>  (1+eps)*X == X), float4 vectorized copy.
// Regular (RT) stores: h0 is immediately re-read/atomically updated, want L2.
// ---------------------------------------------------------------------------
__global__ __launch_bounds__(256) void gin_copy(const float4* __restrict__ X4,
                                                float4* __restrict__ h04,
                                                int n4) {
    int i = blockIdx.x * 256 + threadIdx.x;
    if (i < n4) h04[i] = X4[i];
}

// ---------------------------------------------------------------------------
// Kernel 2: scatter-add  h0[dst[e]] += X[src[e]]
// One wave32 per edge; lane l handles columns 4l..4l+3 via one b128 load and
// 4 global_atomic_add_f32 (no return). Edge indices are wave-uniform ->
// readfirstlane so they become scalar (SMEM) loads and SGPR base addresses.
// X and h0 are L2-resident (51 MB each vs 192 MB L2), so the ~1.6 GB of
// gather+atomic traffic stays on-die instead of hitting HBM.
// ---------------------------------------------------------------------------
__global__ __launch_bounds__(256) void gin_scatter(const float* __restrict__ X,
                                                   const int* __restrict__ esrc,
                                                   const int* __restrict__ edst,
                                                   float* __restrict__ h0,
                                                   int E) {
    int e = blockIdx.x * WAVES_PER_BLOCK + (threadIdx.x >> 5);
    if (e >= E) return;
    int lane = threadIdx.x & 31;
    int s = __builtin_amdgcn_readfirstlane(esrc[e]);
    int d = __builtin_amdgcn_readfirstlane(edst[e]);
    float4 v = ((const float4*)(X + (size_t)s * D))[lane];
    float* hp = h0 + (size_t)d * D + lane * 4;
#if defined(__HIP_PLATFORM_AMD__)
    unsafeAtomicAdd(hp + 0, v.x);
    unsafeAtomicAdd(hp + 1, v.y);
    unsafeAtomicAdd(hp + 2, v.z);
    unsafeAtomicAdd(hp + 3, v.w);
#else
    atomicAdd(hp + 0, v.x);
    atomicAdd(hp + 1, v.y);
    atomicAdd(hp + 2, v.z);
    atomicAdd(hp + 3, v.w);
#endif
}

// ---------------------------------------------------------------------------
// Kernel 3/4: out = act(A @ W + b), A is n_rows x 128, W is 128 x 128.
// Block = 256 threads = 8 waves. W staged in LDS as paired rows:
//   sP[k>>1][col] = { W[k][col], W[k+1][col] }   (k even)  -> exactly 64 KB,
// so each WMMA B-fragment (K = 4kk+2hi, 4kk+2hi+1 ; N = col) is a single
// ds_load_b64 (compiler fuses pairs into ds_load_2addr_b64).
// Each wave computes one 16-row x 128-col tile: 8 v8f accumulators,
// 32 k-steps x 8 col-tiles = 256 v_wmma_f32_16x16x4_f32 (exact fp32 math,
// matching the fp32 reference; this problem is aggregation-bound, so fp32
// WMMA throughput is more than sufficient for the 6.6 GFLOP of GEMM).
// In-place (out == A) is safe: a tile's rows are fully read (k-loop) before
// any store, and each row block is owned by exactly one wave.
// NT_STORE=true for the final layer: 'out' is write-once, keep it from
// evicting the L2-resident h0/X working set.
// ---------------------------------------------------------------------------
template <bool RELU, bool NT_STORE>
__global__ __launch_bounds__(256) void gin_gemm(const float* __restrict__ A,
                                                const float* __restrict__ W,
                                                const float* __restrict__ bias,
                                                float* __restrict__ out,
                                                int n_rows) {
    __shared__ v2f sP[64 * D];  // 64 row-pairs x 128 cols x 8B = 65536 B

    // Cooperative stage of W into paired-row LDS layout.
    for (int idx = threadIdx.x; idx < 64 * D; idx += 256) {
        int p = idx >> 7;        // row pair: rows 2p, 2p+1
        int c = idx & (D - 1);   // column
        v2f w;
        w.x = W[(2 * p) * D + c];
        w.y = W[(2 * p + 1) * D + c];
        sP[idx] = w;
    }
    __syncthreads();

    int wave = threadIdx.x >> 5;
    int lane = threadIdx.x & 31;
    int tile = blockIdx.x * WAVES_PER_BLOCK + wave;  // 16-row tile index
    int ntiles = (n_rows + 15) >> 4;
    if (tile >= ntiles) return;  // wave-uniform: EXEC all-1s inside

    int row0 = tile * 16;
    int m  = lane & 15;  // A: row within tile / B,D: column within col-tile
    int hi = lane >> 4;  // half-wave select

    v8f acc[8];
#pragma unroll
    for (int ct = 0; ct < 8; ++ct) acc[ct] = {};

    // A fragment base: element [row0+m][2*hi], clamped for (generic) ragged N
    int rowA = row0 + m;
    if (rowA >= n_rows) rowA = n_rows - 1;
    const float* Arow = A + (size_t)rowA * D + 2 * hi;

#pragma unroll 4
    for (int kk = 0; kk < 32; ++kk) {
        // A(16x4 f32): lane -> { a.x = A[m][4kk+2hi], a.y = A[m][4kk+2hi+1] }
        v2f a = *(const v2f*)(Arow + 4 * kk);
        int bidx = (2 * kk + hi) * D + m;  // pair (4kk+2hi, +1), col base m
#pragma unroll
        for (int ct = 0; ct < 8; ++ct) {
            // B(4x16): { b.x = W[4kk+2hi][16ct+m], b.y = W[4kk+2hi+1][16ct+m] }
            v2f b = sP[bidx + ct * 16];
            acc[ct] = __builtin_amdgcn_wmma_f32_16x16x4_f32(
                false, a, false, b, (short)0, acc[ct], false, false);
        }
    }

    // Epilogue: D layout row = r + 8*hi, col = 16*ct + m
#pragma unroll
    for (int ct = 0; ct < 8; ++ct) {
        int col = ct * 16 + m;
        float bv = bias[col];
#pragma unroll
        for (int r = 0; r < 8; ++r) {
            int row = row0 + r + 8 * hi;
            if (row < n_rows) {
                float v = acc[ct][r] + bv;
                if (RELU) v = fmaxf(v, 0.0f);
                float* p = out + (size_t)row * D + col;
                if (NT_STORE)
                    __builtin_nontemporal_store(v, p);
                else
                    *p = v;
            }
        }
    }
}

// ---------------------------------------------------------------------------
// Launch: h0 = X ; h0 += scatter(X) ; h0 = relu(h0@W1+b1) ; out = h0@W2+b2
// ---------------------------------------------------------------------------
extern "C" void kernel_launch(void* const* d_in, const int* in_sizes, int n_in,
                              void* d_out, int out_size, void* d_ws, size_t ws_size,
                              hipStream_t stream) {
    const float* X  = (const float*)d_in[0];
    const float* W1 = (const float*)d_in[1];
    const float* b1 = (const float*)d_in[2];
    const float* W2 = (const float*)d_in[3];
    const float* b2 = (const float*)d_in[4];
    const int* esrc = (const int*)d_in[5];
    const int* edst = (const int*)d_in[6];

    int N = in_sizes[0] / D;   // 100000
    int E = in_sizes[5];       // 1600000

    float* h0  = (float*)d_ws;        // N*D floats = 51.2 MB scratch
    float* out = (float*)d_out;

    int n4 = N * (D / 4);
    gin_copy<<<(n4 + 255) / 256, 256, 0, stream>>>(
        (const float4*)X, (float4*)h0, n4);

    gin_scatter<<<(E + WAVES_PER_BLOCK - 1) / WAVES_PER_BLOCK, 256, 0, stream>>>(
        X, esrc, edst, h0, E);

    int ntiles  = (N + 15) / 16;
    int nblocks = (ntiles + WAVES_PER_BLOCK - 1) / WAVES_PER_BLOCK;
    gin_gemm<true, false><<<nblocks, 256, 0, stream>>>(h0, W1, b1, h0, N);
    gin_gemm<false, true><<<nblocks, 256, 0, stream>>>(h0, W2, b2, out, N);
}